// CrossViewAttention_5592047419813
// MI455X (gfx1250) — compile-verified
//
#include <hip/hip_runtime.h>

#define D 64
#define ALPHA_LEAKY 0.2f
#define EPS_F 1e-10f

typedef __attribute__((ext_vector_type(2))) float v2f;
typedef __attribute__((ext_vector_type(8))) float v8f;

// ---- monotonic float <-> uint encoding for atomicMax on floats ----
__device__ __forceinline__ unsigned f2ord(float f) {
  unsigned b = __float_as_uint(f);
  return (b & 0x80000000u) ? ~b : (b | 0x80000000u);
}
__device__ __forceinline__ float ord2f(unsigned u) {
  unsigned b = (u & 0x80000000u) ? (u & 0x7FFFFFFFu) : ~u;
  return __uint_as_float(b);
}

// ---- 1) fold proj_w into the two attention vectors: v = W^T wq, u = W^T wk ----
__global__ void k_prep(const float* __restrict__ proj_w, const float* __restrict__ proj_b,
                       const float* __restrict__ attend_w, const float* __restrict__ attend_b,
                       float* __restrict__ vu) {
  int d = threadIdx.x;
  if (d < D) {
    float v = 0.f, u = 0.f;
    for (int j = 0; j < D; ++j) {
      float w = proj_w[j * D + d];
      v += w * attend_w[j];
      u += w * attend_w[D + j];
    }
    vu[d] = v;
    vu[D + d] = u;
  }
  if (d == 0) {
    float cq = 0.f, ck = 0.f;
    for (int j = 0; j < D; ++j) { cq += proj_b[j] * attend_w[j]; ck += proj_b[j] * attend_w[D + j]; }
    vu[2 * D]     = cq;
    vu[2 * D + 1] = ck;
    vu[2 * D + 2] = attend_b[0];
  }
}

// ---- 2) per-node scalar scores sq[n] = q_n . v + cq ; sk[n] = k_n . u + ck ----
__global__ __launch_bounds__(256) void k_scores(const float* __restrict__ qn, const float* __restrict__ kn,
                                                const float* __restrict__ vu,
                                                float* __restrict__ sq, float* __restrict__ sk,
                                                int Nq, int Nk) {
  int n = blockIdx.x * blockDim.x + threadIdx.x;
  if (n < Nq) {
    const float4* r = (const float4*)(qn + (size_t)n * D);
    const float4* v = (const float4*)(vu);
    float s = vu[2 * D];
#pragma unroll
    for (int i = 0; i < D / 4; ++i) {
      float4 a = r[i], b = v[i];
      s += a.x * b.x + a.y * b.y + a.z * b.z + a.w * b.w;
    }
    sq[n] = s;
  }
  if (n < Nk) {
    const float4* r = (const float4*)(kn + (size_t)n * D);
    const float4* v = (const float4*)(vu + D);
    float s = vu[2 * D + 1];
#pragma unroll
    for (int i = 0; i < D / 4; ++i) {
      float4 a = r[i], b = v[i];
      s += a.x * b.x + a.y * b.y + a.z * b.z + a.w * b.w;
    }
    sk[n] = s;
  }
}

// ---- 3) kv_proj = K @ W^T + b via V_WMMA_F32_16X16X4_F32 (wave32, 16 rows x 64 cols per wave) ----
// A (16x4 f32): lane l holds A[M=l&15][K = k4 + 2*(l>>4) + {0,1}] in {v0,v1}
// B (4x16 f32): lane l holds B[K = k4 + 2*(l>>4) + {0,1}][N=l&15]  -> B[d][j] = proj_w[j][d]
// C/D (16x16):  vgpr r, lane l -> [M = r + 8*(l>>4)][N = l&15]
__global__ __launch_bounds__(256) void k_proj_wmma(const float* __restrict__ kn,
                                                   const float* __restrict__ proj_w,
                                                   const float* __restrict__ proj_b,
                                                   float* __restrict__ kvp, int Nk) {
  int wave = (int)((blockIdx.x * blockDim.x + threadIdx.x) >> 5);
  int lane = threadIdx.x & 31;
  int ntiles = Nk >> 4;
  if (wave >= ntiles) return;

  int m  = lane & 15;
  int hi = lane >> 4;  // 0 or 1
  const float* arow = kn + ((size_t)(wave * 16 + m)) * D + 2 * hi;
  const float* bmat = proj_w + 2 * hi;

  v8f acc0 = {}, acc1 = {}, acc2 = {}, acc3 = {};
#pragma unroll
  for (int k4 = 0; k4 < D; k4 += 4) {
    v2f a;  a.x = arow[k4];  a.y = arow[k4 + 1];
    v2f b0; b0.x = bmat[(0 * 16 + m) * D + k4]; b0.y = bmat[(0 * 16 + m) * D + k4 + 1];
    v2f b1; b1.x = bmat[(1 * 16 + m) * D + k4]; b1.y = bmat[(1 * 16 + m) * D + k4 + 1];
    v2f b2; b2.x = bmat[(2 * 16 + m) * D + k4]; b2.y = bmat[(2 * 16 + m) * D + k4 + 1];
    v2f b3; b3.x = bmat[(3 * 16 + m) * D + k4]; b3.y = bmat[(3 * 16 + m) * D + k4 + 1];
    acc0 = __builtin_amdgcn_wmma_f32_16x16x4_f32(false, a, false, b0, (short)0, acc0, false, false);
    acc1 = __builtin_amdgcn_wmma_f32_16x16x4_f32(false, a, false, b1, (short)0, acc1, false, false);
    acc2 = __builtin_amdgcn_wmma_f32_16x16x4_f32(false, a, false, b2, (short)0, acc2, false, false);
    acc3 = __builtin_amdgcn_wmma_f32_16x16x4_f32(false, a, false, b3, (short)0, acc3, false, false);
  }

  size_t rowbase = (size_t)(wave * 16 + 8 * hi) * D + m;
  float pb0 = proj_b[0 * 16 + m], pb1 = proj_b[1 * 16 + m];
  float pb2 = proj_b[2 * 16 + m], pb3 = proj_b[3 * 16 + m];
#pragma unroll
  for (int r = 0; r < 8; ++r) {
    size_t off = rowbase + (size_t)r * D;
    kvp[off + 0 * 16] = acc0[r] + pb0;
    kvp[off + 1 * 16] = acc1[r] + pb1;
    kvp[off + 2 * 16] = acc2[r] + pb2;
    kvp[off + 3 * 16] = acc3[r] + pb3;
  }
}

// ---- 4) init: zero output accumulator + segment reducers ----
__global__ void k_init(float* __restrict__ out, float* __restrict__ esum,
                       unsigned* __restrict__ emax, int Nout, int Nq) {
  int i = blockIdx.x * blockDim.x + threadIdx.x;
  if (i < Nout) out[i] = 0.f;
  if (i < Nq) { esum[i] = 0.f; emax[i] = 0u; }  // ord 0 == -NaN == lowest
}

// ---- 5) per-edge score + segment max (hardware u32 atomicMax on ordered encoding) ----
__global__ __launch_bounds__(256) void k_edge_max(const int* __restrict__ qi, const int* __restrict__ ki,
                                                  const float* __restrict__ sq, const float* __restrict__ sk,
                                                  const float* __restrict__ vu,
                                                  float* __restrict__ ew, unsigned* __restrict__ emax, int E) {
  int i = blockIdx.x * blockDim.x + threadIdx.x;
  if (i >= E) return;
  int q = qi[i], k = ki[i];
  float e = sq[q] + sk[k] + vu[2 * D + 2];
  e = e > 0.f ? e : ALPHA_LEAKY * e;
  ew[i] = e;
  atomicMax(&emax[q], f2ord(e));
}

// ---- 6) per-edge exp(e - max) + segment sum (global_atomic_add_f32) ----
__global__ __launch_bounds__(256) void k_edge_sum(const int* __restrict__ qi,
                                                  float* __restrict__ ew, const unsigned* __restrict__ emax,
                                                  float* __restrict__ esum, int E) {
  int i = blockIdx.x * blockDim.x + threadIdx.x;
  if (i >= E) return;
  int q = qi[i];
  float w = expf(ew[i] - ord2f(emax[q]));
  ew[i] = w;
  atomicAdd(&esum[q], w);
}

// ---- 7) wave-per-edge weighted scatter: out[q] += attn * kv_proj[k]  (64 floats = 32 lanes x float2) ----
__global__ __launch_bounds__(256) void k_scatter(const int* __restrict__ qi, const int* __restrict__ ki,
                                                 const float* __restrict__ ew, const float* __restrict__ esum,
                                                 const float* __restrict__ kvp, float* __restrict__ out, int E) {
  int edge = (int)((blockIdx.x * blockDim.x + threadIdx.x) >> 5);
  int lane = threadIdx.x & 31;
  if (edge >= E) return;
  int q = qi[edge], k = ki[edge];
  float attn = ew[edge] / (esum[q] + EPS_F);
  const float2* src = (const float2*)(kvp + (size_t)k * D);
  float2 v = src[lane];
  float* dst = out + (size_t)q * D + 2 * lane;
  atomicAdd(dst + 0, v.x * attn);
  atomicAdd(dst + 1, v.y * attn);
}

extern "C" void kernel_launch(void* const* d_in, const int* in_sizes, int n_in,
                              void* d_out, int out_size, void* d_ws, size_t ws_size,
                              hipStream_t stream) {
  const float* qn  = (const float*)d_in[0];
  const float* kn  = (const float*)d_in[1];
  const int*   ei  = (const int*)d_in[2];
  const float* pw  = (const float*)d_in[3];
  const float* pb  = (const float*)d_in[4];
  const float* aw  = (const float*)d_in[5];
  const float* ab  = (const float*)d_in[6];
  float* out = (float*)d_out;

  const int Nq = in_sizes[0] / D;
  const int Nk = in_sizes[1] / D;
  const int E  = in_sizes[2] / 2;
  const int* qi = ei;
  const int* ki = ei + E;

  // workspace layout (all 4-byte elements)
  float*    kvp  = (float*)d_ws;            // Nk*D
  float*    sq   = kvp + (size_t)Nk * D;    // Nq
  float*    sk   = sq + Nq;                 // Nk
  unsigned* emax = (unsigned*)(sk + Nk);    // Nq
  float*    esum = (float*)(emax + Nq);     // Nq
  float*    ew   = esum + Nq;               // E
  float*    vu   = ew + E;                  // 2*D + 3

  k_prep<<<1, D, 0, stream>>>(pw, pb, aw, ab, vu);

  int nmax = Nq > Nk ? Nq : Nk;
  k_scores<<<(nmax + 255) / 256, 256, 0, stream>>>(qn, kn, vu, sq, sk, Nq, Nk);

  int ntiles = Nk / 16;                 // Nk = 100000, multiple of 16
  k_proj_wmma<<<(ntiles + 7) / 8, 256, 0, stream>>>(kn, pw, pb, kvp, Nk);

  int nout = Nq * D;
  k_init<<<(nout + 255) / 256, 256, 0, stream>>>(out, esum, emax, nout, Nq);

  k_edge_max<<<(E + 255) / 256, 256, 0, stream>>>(qi, ki, sq, sk, vu, ew, emax, E);
  k_edge_sum<<<(E + 255) / 256, 256, 0, stream>>>(qi, ew, emax, esum, E);
  k_scatter<<<((size_t)E * 32 + 255) / 256, 256, 0, stream>>>(qi, ki, ew, esum, kvp, out, E);
}